// Coma_14559939133576
// MI455X (gfx1250) — compile-verified
//
#include <hip/hip_runtime.h>
#include <math.h>

typedef __attribute__((ext_vector_type(2))) float v2f;
typedef __attribute__((ext_vector_type(8))) float v8f;

static inline unsigned cdiv(long n, int b) { return (unsigned)((n + (long)b - 1) / b); }

// ---------------- elementwise ----------------
__global__ void k_fill(float* p, long n, float v) {
  long i = (long)blockIdx.x * blockDim.x + threadIdx.x;
  if (i < n) p[i] = v;
}
__global__ void k_fill_bias(float* out, const float* bias, long rows, int F) {
  long i = (long)blockIdx.x * blockDim.x + threadIdx.x;
  if (i < rows * (long)F) out[i] = bias ? bias[i % F] : 0.0f;
}
__global__ void k_relu(float* p, long n) {
  long i = (long)blockIdx.x * blockDim.x + threadIdx.x;
  if (i < n) p[i] = fmaxf(p[i], 0.0f);
}
__global__ void k_axpby(float* z, const float* x, const float* y, float a, float b, long n) {
  long i = (long)blockIdx.x * blockDim.x + threadIdx.x;
  if (i < n) z[i] = a * x[i] + b * y[i];
}

// ---------------- graph setup ----------------
__global__ void k_deg(const int* row, float* deg, int E) {
  int e = blockIdx.x * blockDim.x + threadIdx.x;
  if (e < E) atomicAdd(&deg[row[e]], 1.0f);
}
__global__ void k_dinv(float* deg, int N) {
  int i = blockIdx.x * blockDim.x + threadIdx.x;
  if (i < N) { float d = deg[i]; deg[i] = (d > 0.0f) ? (1.0f / sqrtf(d)) : 0.0f; }
}
__global__ void k_norm(const int* row, const int* col, const float* dinv, float* nrm, int E) {
  int e = blockIdx.x * blockDim.x + threadIdx.x;
  if (e < E) nrm[e] = dinv[row[e]] * dinv[col[e]];
}
__global__ void k_adjset(const int* row, const int* col, float* adj, int E, int N) {
  int e = blockIdx.x * blockDim.x + threadIdx.x;
  if (e < E) adj[(long)row[e] * N + col[e]] = 1.0f;
}

// sparse normalized propagation: out[b,row,f] += nrm[e]*h[b,col,f]
__global__ void k_spprop(float* out, const float* h, const int* row, const int* col,
                         const float* nrm, int Bc, int E, int N, int F) {
  long idx = (long)blockIdx.x * blockDim.x + threadIdx.x;
  long tot = (long)Bc * E;
  if (idx >= tot) return;
  int b = (int)(idx / E);
  int e = (int)(idx % E);
  float nv = nrm[e];
  const float* src = h + ((long)b * N + col[e]) * F;
  float* dst = out + ((long)b * N + row[e]) * F;
  for (int f = 0; f < F; ++f) atomicAdd(&dst[f], nv * src[f]);
}

// row softmax (in place) + entropy accumulation
__global__ void k_softmax_ent(float* s, int O, float* ent, float entScale) {
  long rowi = blockIdx.x;
  float* p = s + rowi * (long)O;
  __shared__ float red[256];
  int tid = threadIdx.x;
  float m = -3.4e38f;
  for (int o = tid; o < O; o += 256) m = fmaxf(m, p[o]);
  red[tid] = m; __syncthreads();
  for (int st = 128; st > 0; st >>= 1) { if (tid < st) red[tid] = fmaxf(red[tid], red[tid + st]); __syncthreads(); }
  m = red[0]; __syncthreads();
  float sum = 0.0f;
  for (int o = tid; o < O; o += 256) sum += expf(p[o] - m);
  red[tid] = sum; __syncthreads();
  for (int st = 128; st > 0; st >>= 1) { if (tid < st) red[tid] += red[tid + st]; __syncthreads(); }
  sum = red[0]; __syncthreads();
  float inv = 1.0f / sum;
  float eloc = 0.0f;
  for (int o = tid; o < O; o += 256) {
    float v = expf(p[o] - m) * inv;
    p[o] = v;
    eloc += -v * logf(v + 1e-15f);
  }
  red[tid] = eloc; __syncthreads();
  for (int st = 128; st > 0; st >>= 1) { if (tid < st) red[tid] += red[tid + st]; __syncthreads(); }
  if (tid == 0) atomicAdd(ent, red[0] * entScale);
}

// ---------------- generic batched WMMA GEMM (f32, 16x16x4) ----------------
// C[b] = alpha * op(A[b]) @ op(B[b]) + beta * C[b]; row-major, batch strides (0 = broadcast).
// TA/TB compile-time. Full interior tiles take a software-pipelined, branch-free loop
// (double-buffered fragments so loads for k+4 are in flight during the wmma for k);
// edge tiles use clamped addresses + v_cndmask masks (no exec divergence at the wmma).
template <int TA, int TB>
__global__ void k_gemm_t(const float* __restrict__ A, const float* __restrict__ Bm,
                         float* __restrict__ C,
                         int M, int N, int Kd, int lda, int ldb, int ldc,
                         long sA, long sB, long sC, int batches,
                         float alpha, float beta) {
  long gtid = (long)blockIdx.x * blockDim.x + threadIdx.x;
  int wv = (int)(gtid >> 5);
  int lane = threadIdx.x & 31;
  int tN = (N + 15) >> 4, tM = (M + 15) >> 4;
  long total = (long)batches * tM * tN;
  if (wv >= total) return;
  int b = wv / (tM * tN);
  int t = wv % (tM * tN);
  int tm = t / tN, tn = t % tN;
  const float* Ab = A + (long)b * sA;
  const float* Bb = Bm + (long)b * sB;
  float* Cb = C + (long)b * sC;
  int hlf = lane >> 4;
  int l = lane & 15;
  int am = tm * 16 + l;
  int bn = tn * 16 + l;
  bool full = (tm * 16 + 16 <= M) && (tn * 16 + 16 <= N);  // wave-uniform
  int Kmain = Kd & ~3;
  v8f acc = {0.f, 0.f, 0.f, 0.f, 0.f, 0.f, 0.f, 0.f};

  if (full) {
    const float* pa = TA ? (Ab + am) : (Ab + (long)am * lda);
    const float* pb = TB ? (Bb + (long)bn * ldb) : (Bb + bn);
    if (Kmain > 0) {
      int ka = 2 * hlf;
      v2f a0, b0;
      a0[0] = TA ? pa[(long)ka * lda]       : pa[ka];
      a0[1] = TA ? pa[(long)(ka + 1) * lda] : pa[ka + 1];
      b0[0] = TB ? pb[ka]     : pb[(long)ka * ldb];
      b0[1] = TB ? pb[ka + 1] : pb[(long)(ka + 1) * ldb];
      for (int k0 = 0; k0 + 4 < Kmain; k0 += 4) {
        int kn = k0 + 4 + 2 * hlf;
        v2f a1, b1;
        a1[0] = TA ? pa[(long)kn * lda]       : pa[kn];
        a1[1] = TA ? pa[(long)(kn + 1) * lda] : pa[kn + 1];
        b1[0] = TB ? pb[kn]     : pb[(long)kn * ldb];
        b1[1] = TB ? pb[kn + 1] : pb[(long)(kn + 1) * ldb];
        acc = __builtin_amdgcn_wmma_f32_16x16x4_f32(false, a0, false, b0, (short)0, acc, false, false);
        a0 = a1; b0 = b1;
      }
      acc = __builtin_amdgcn_wmma_f32_16x16x4_f32(false, a0, false, b0, (short)0, acc, false, false);
    }
    if (Kmain < Kd) {
      int ka = Kmain + 2 * hlf;
      int k0c = ka < Kd ? ka : Kd - 1;
      int k1c = (ka + 1) < Kd ? (ka + 1) : Kd - 1;
      float a0 = TA ? pa[(long)k0c * lda] : pa[k0c];
      float a1 = TA ? pa[(long)k1c * lda] : pa[k1c];
      float b0 = TB ? pb[k0c] : pb[(long)k0c * ldb];
      float b1 = TB ? pb[k1c] : pb[(long)k1c * ldb];
      v2f af, bf;
      af[0] = (ka < Kd) ? a0 : 0.f;
      af[1] = (ka + 1 < Kd) ? a1 : 0.f;
      bf[0] = (ka < Kd) ? b0 : 0.f;
      bf[1] = (ka + 1 < Kd) ? b1 : 0.f;
      acc = __builtin_amdgcn_wmma_f32_16x16x4_f32(false, af, false, bf, (short)0, acc, false, false);
    }
    int mb = tm * 16 + hlf * 8;
    int n = tn * 16 + l;
    for (int j = 0; j < 8; ++j) {
      long ci = (long)(mb + j) * ldc + n;
      float v = alpha * acc[j];
      if (beta != 0.0f) v += beta * Cb[ci];
      Cb[ci] = v;
    }
  } else {
    int amc = am < M ? am : M - 1;
    int bnc = bn < N ? bn : N - 1;
    bool amv = am < M, bnv = bn < N;
    const float* pa = TA ? (Ab + amc) : (Ab + (long)amc * lda);
    const float* pb = TB ? (Bb + (long)bnc * ldb) : (Bb + bnc);
    for (int k0 = 0; k0 < Kd; k0 += 4) {
      int ka = k0 + 2 * hlf;
      int k0c = ka < Kd ? ka : Kd - 1;
      int k1c = (ka + 1) < Kd ? (ka + 1) : Kd - 1;
      float a0 = TA ? pa[(long)k0c * lda] : pa[k0c];
      float a1 = TA ? pa[(long)k1c * lda] : pa[k1c];
      float b0 = TB ? pb[k0c] : pb[(long)k0c * ldb];
      float b1 = TB ? pb[k1c] : pb[(long)k1c * ldb];
      v2f af, bf;
      af[0] = (amv && ka < Kd) ? a0 : 0.f;
      af[1] = (amv && ka + 1 < Kd) ? a1 : 0.f;
      bf[0] = (bnv && ka < Kd) ? b0 : 0.f;
      bf[1] = (bnv && ka + 1 < Kd) ? b1 : 0.f;
      acc = __builtin_amdgcn_wmma_f32_16x16x4_f32(false, af, false, bf, (short)0, acc, false, false);
    }
    int mb = tm * 16 + hlf * 8;
    int n = tn * 16 + l;
    for (int j = 0; j < 8; ++j) {
      int m = mb + j;
      if (m < M && n < N) {
        long ci = (long)m * ldc + n;
        float v = alpha * acc[j];
        if (beta != 0.0f) v += beta * Cb[ci];
        Cb[ci] = v;
      }
    }
  }
}

// fused link loss: accum += sum over (adj[b] - S[b] S[b]^T)^2 ; sst never materialized.
// Both fragments are row-contiguous reads of S; main loop is software-pipelined.
__global__ void k_link_sst(const float* __restrict__ S, const float* __restrict__ Adj,
                           int I, int O, long sS, long sAdj, int batches, float* accum) {
  long gtid = (long)blockIdx.x * blockDim.x + threadIdx.x;
  int wv = (int)(gtid >> 5);
  int lane = threadIdx.x & 31;
  int tI = (I + 15) >> 4;
  long total = (long)batches * tI * tI;
  if (wv >= total) return;
  int b = wv / (tI * tI);
  int t = wv % (tI * tI);
  int tm = t / tI, tn = t % tI;
  const float* Sb = S + (long)b * sS;
  const float* Ab = Adj + (long)b * sAdj;
  int hlf = lane >> 4, l = lane & 15;
  int am = tm * 16 + l;
  int bn = tn * 16 + l;
  bool full = (tm * 16 + 16 <= I) && (tn * 16 + 16 <= I);  // wave-uniform
  int Kmain = O & ~3;
  v8f acc = {0.f, 0.f, 0.f, 0.f, 0.f, 0.f, 0.f, 0.f};
  int amc = am < I ? am : I - 1;
  int bnc = bn < I ? bn : I - 1;
  const float* pa = Sb + (long)amc * O;
  const float* pb = Sb + (long)bnc * O;

  if (Kmain > 0) {
    int ka = 2 * hlf;
    v2f a0, b0;
    a0[0] = pa[ka];     a0[1] = pa[ka + 1];
    b0[0] = pb[ka];     b0[1] = pb[ka + 1];
    for (int k0 = 0; k0 + 4 < Kmain; k0 += 4) {
      int kn = k0 + 4 + 2 * hlf;
      v2f a1, b1;
      a1[0] = pa[kn];     a1[1] = pa[kn + 1];
      b1[0] = pb[kn];     b1[1] = pb[kn + 1];
      acc = __builtin_amdgcn_wmma_f32_16x16x4_f32(false, a0, false, b0, (short)0, acc, false, false);
      a0 = a1; b0 = b1;
    }
    acc = __builtin_amdgcn_wmma_f32_16x16x4_f32(false, a0, false, b0, (short)0, acc, false, false);
  }
  if (Kmain < O) {
    int ka = Kmain + 2 * hlf;
    int k0c = ka < O ? ka : O - 1;
    int k1c = (ka + 1) < O ? (ka + 1) : O - 1;
    float a0 = pa[k0c], a1 = pa[k1c];
    float b0 = pb[k0c], b1 = pb[k1c];
    v2f af, bf;
    af[0] = (ka < O) ? a0 : 0.f;
    af[1] = (ka + 1 < O) ? a1 : 0.f;
    bf[0] = (ka < O) ? b0 : 0.f;
    bf[1] = (ka + 1 < O) ? b1 : 0.f;
    acc = __builtin_amdgcn_wmma_f32_16x16x4_f32(false, af, false, bf, (short)0, acc, false, false);
  }
  int mb = tm * 16 + hlf * 8;
  int n = tn * 16 + l;
  float part = 0.0f;
  if (full) {
    for (int j = 0; j < 8; ++j) {
      float d = Ab[(long)(mb + j) * I + n] - acc[j];
      part += d * d;
    }
  } else {
    for (int j = 0; j < 8; ++j) {
      int m = mb + j;
      if (m < I && n < I) {
        float d = Ab[(long)m * I + n] - acc[j];
        part += d * d;
      }
    }
  }
  for (int off = 16; off > 0; off >>= 1) part += __shfl_xor(part, off, 32);
  if (lane == 0) atomicAdd(accum, part);
}

__global__ void k_link_fin(float* link, float* accum, float invSize) {
  *link += sqrtf(*accum) * invSize;
  *accum = 0.0f;
}
__global__ void k_emit(float* dout, const float* link, const float* ent, long off) {
  dout[off] = *link;
  dout[off + 1] = *ent;
}

// ---------------- host orchestration ----------------
extern "C" void kernel_launch(void* const* d_in, const int* in_sizes, int n_in,
                              void* d_out, int out_size, void* d_ws, size_t ws_size,
                              hipStream_t stream) {
  (void)in_sizes; (void)out_size;
  if (n_in < 37) return;
  const int Bc = 4, Kc = 6, N0 = 2000, E = 24000;

  const float* x = (const float*)d_in[0];
  const int* ei = (const int*)d_in[1];
  const int* rowp = ei;
  const int* colp = ei + E;
  const float* enc_w[4] = {(const float*)d_in[2], (const float*)d_in[3], (const float*)d_in[4], (const float*)d_in[5]};
  const float* enc_b[4] = {(const float*)d_in[6], (const float*)d_in[7], (const float*)d_in[8], (const float*)d_in[9]};
  const float* pool_w[4] = {(const float*)d_in[10], (const float*)d_in[11], (const float*)d_in[12], (const float*)d_in[13]};
  const float* pool_b[4] = {(const float*)d_in[14], (const float*)d_in[15], (const float*)d_in[16], (const float*)d_in[17]};
  const float* enc_lin_w = (const float*)d_in[18];
  const float* enc_lin_b = (const float*)d_in[19];
  const float* dec_lin_w = (const float*)d_in[20];
  const float* dec_lin_b = (const float*)d_in[21];
  const float* unp_w[4] = {(const float*)d_in[22], (const float*)d_in[23], (const float*)d_in[24], (const float*)d_in[25]};
  const float* unp_b[4] = {(const float*)d_in[26], (const float*)d_in[27], (const float*)d_in[28], (const float*)d_in[29]};
  const float* dec_w[4] = {(const float*)d_in[30], (const float*)d_in[31], (const float*)d_in[32], (const float*)d_in[33]};
  const float* dec_b[3] = {(const float*)d_in[34], (const float*)d_in[35], (const float*)d_in[36]};

  float* ws = (float*)d_ws;
  size_t off = 0;
  auto alloc = [&](size_t n) { size_t r = off; off += (n + 63) & ~(size_t)63; return ws + r; };
  float* scal = alloc(64);            // [0]=link [1]=ent [2]=accum
  float* link = scal, *ent = scal + 1, *accum = scal + 2;
  float* dinv = alloc(2048);
  float* nrm  = alloc(E);
  float* adj0 = alloc((size_t)N0 * N0);       // 4M
  float* bufS = alloc(4000000);               // s_logits / s
  float* bufT = alloc(4000000);               // s^T @ adj temp
  float* adjA = alloc(1000000);
  float* adjB = alloc(1000000);
  float* hA = alloc(131072);
  float* hB = alloc(131072);
  float* cX = alloc(131072);
  float* cY = alloc(131072);
  float* cZ = alloc(131072);
  float* cT = alloc(131072);
  float* zb = alloc(2048);
  if (off * sizeof(float) > ws_size) return;

  auto fill = [&](float* p, long n, float v) { k_fill<<<cdiv(n, 256), 256, 0, stream>>>(p, n, v); };
  auto fillb = [&](float* o, const float* b, long rows, int F) {
    k_fill_bias<<<cdiv(rows * (long)F, 256), 256, 0, stream>>>(o, b, rows, F);
  };
  auto relu = [&](float* p, long n) { k_relu<<<cdiv(n, 256), 256, 0, stream>>>(p, n); };
  auto gemm = [&](const float* A, const float* B_, float* C, int M, int N, int Kd,
                  int lda, int ldb, int ldc, long sA, long sB, long sC, int nb,
                  int tA, int tB, float al, float be) {
    long waves = (long)nb * ((M + 15) / 16) * ((N + 15) / 16);
    unsigned blocks = cdiv(waves * 32, 256);
    if (!tA && !tB)
      k_gemm_t<0, 0><<<blocks, 256, 0, stream>>>(A, B_, C, M, N, Kd, lda, ldb, ldc, sA, sB, sC, nb, al, be);
    else if (tA && !tB)
      k_gemm_t<1, 0><<<blocks, 256, 0, stream>>>(A, B_, C, M, N, Kd, lda, ldb, ldc, sA, sB, sC, nb, al, be);
    else if (!tA && tB)
      k_gemm_t<0, 1><<<blocks, 256, 0, stream>>>(A, B_, C, M, N, Kd, lda, ldb, ldc, sA, sB, sC, nb, al, be);
    else
      k_gemm_t<1, 1><<<blocks, 256, 0, stream>>>(A, B_, C, M, N, Kd, lda, ldb, ldc, sA, sB, sC, nb, al, be);
  };
  auto spprop = [&](float* outp, const float* hin, int F) {
    fill(outp, (long)Bc * N0 * F, 0.0f);
    long tot = (long)Bc * E;
    k_spprop<<<cdiv(tot, 256), 256, 0, stream>>>(outp, hin, rowp, colp, nrm, Bc, E, N0, F);
  };

  // Chebyshev conv, dense adjacency (batched, sAdj=0 means broadcast)
  auto cheb_dense = [&](const float* hin, int I, int Fin, int Fout,
                        const float* adj, long sAdj,
                        const float* w, const float* bias, float* out) {
    fillb(out, bias, (long)Bc * I, Fout);
    gemm(hin, w, out, Bc * I, Fout, Fin, Fin, Fout, Fout, 0, 0, 0, 1, 0, 0, 1.f, 1.f);
    gemm(adj, hin, cX, I, Fin, I, I, Fin, Fin, sAdj, (long)I * Fin, (long)I * Fin, Bc, 0, 0, 1.f, 0.f);
    gemm(cX, w + (size_t)Fin * Fout, out, Bc * I, Fout, Fin, Fin, Fout, Fout, 0, 0, 0, 1, 0, 0, 1.f, 1.f);
    const float* p2 = hin; const float* p1 = cX;
    float* rot[3] = {cY, cZ, cX};
    long cnt = (long)Bc * I * Fin;
    for (int k = 2; k < Kc; ++k) {
      float* nx = rot[(k - 2) % 3];
      gemm(adj, p1, cT, I, Fin, I, I, Fin, Fin, sAdj, (long)I * Fin, (long)I * Fin, Bc, 0, 0, 1.f, 0.f);
      k_axpby<<<cdiv(cnt, 256), 256, 0, stream>>>(nx, cT, p2, 2.f, -1.f, cnt);
      gemm(nx, w + (size_t)k * Fin * Fout, out, Bc * I, Fout, Fin, Fin, Fout, Fout, 0, 0, 0, 1, 0, 0, 1.f, 1.f);
      p2 = p1; p1 = nx;
    }
  };
  // Chebyshev conv, sparse template graph (N0 nodes)
  auto cheb_sparse = [&](const float* hin, int Fin, int Fout,
                         const float* w, const float* bias, float* out) {
    fillb(out, bias, (long)Bc * N0, Fout);
    gemm(hin, w, out, Bc * N0, Fout, Fin, Fin, Fout, Fout, 0, 0, 0, 1, 0, 0, 1.f, 1.f);
    spprop(cX, hin, Fin);
    gemm(cX, w + (size_t)Fin * Fout, out, Bc * N0, Fout, Fin, Fin, Fout, Fout, 0, 0, 0, 1, 0, 0, 1.f, 1.f);
    const float* p2 = hin; const float* p1 = cX;
    float* rot[3] = {cY, cZ, cX};
    long cnt = (long)Bc * N0 * Fin;
    for (int k = 2; k < Kc; ++k) {
      float* nx = rot[(k - 2) % 3];
      spprop(cT, p1, Fin);
      k_axpby<<<cdiv(cnt, 256), 256, 0, stream>>>(nx, cT, p2, 2.f, -1.f, cnt);
      gemm(nx, w + (size_t)k * Fin * Fout, out, Bc * N0, Fout, Fin, Fin, Fout, Fout, 0, 0, 0, 1, 0, 0, 1.f, 1.f);
      p2 = p1; p1 = nx;
    }
  };
  // DiffPool: softmax(s)+ent, link (fused sst), hp = s^T h, optionally adj' = s^T adj s
  auto diff_pool = [&](float* s, int I, int O, const float* hin, int F, float* hpOut,
                       const float* adj, long sAdj, float* t1, float* adjOut, bool computeAdj) {
    k_softmax_ent<<<Bc * I, 256, 0, stream>>>(s, O, ent, 1.0f / (float)(Bc * I));
    long waves = (long)Bc * ((I + 15) / 16) * ((I + 15) / 16);
    k_link_sst<<<cdiv(waves * 32, 256), 256, 0, stream>>>(s, adj, I, O, (long)I * O, sAdj, Bc, accum);
    k_link_fin<<<1, 1, 0, stream>>>(link, accum, 1.0f / ((float)Bc * I * I));
    gemm(s, hin, hpOut, O, F, I, O, F, F, (long)I * O, (long)I * F, (long)O * F, Bc, 1, 0, 1.f, 0.f);
    if (computeAdj) {
      gemm(s, adj, t1, O, I, I, O, I, I, (long)I * O, sAdj, (long)O * I, Bc, 1, 0, 1.f, 0.f);
      gemm(t1, s, adjOut, O, O, I, I, O, O, (long)O * I, (long)I * O, (long)O * O, Bc, 0, 0, 1.f, 0.f);
    }
  };

  // ---------- graph normalization + dense template adjacency ----------
  fill(scal, 64, 0.0f);
  fill(dinv, 2048, 0.0f);
  k_deg<<<cdiv(E, 256), 256, 0, stream>>>(rowp, dinv, E);
  k_dinv<<<cdiv(N0, 256), 256, 0, stream>>>(dinv, N0);
  k_norm<<<cdiv(E, 256), 256, 0, stream>>>(rowp, colp, dinv, nrm, E);
  fill(adj0, (long)N0 * N0, 0.0f);
  k_adjset<<<cdiv(E, 256), 256, 0, stream>>>(rowp, colp, adj0, E, N0);

  // ---------- encoder ----------
  cheb_sparse(x, 3, 16, enc_w[0], enc_b[0], hA);            relu(hA, (long)Bc * N0 * 16);
  cheb_sparse(hA, 16, 500, pool_w[0], pool_b[0], bufS);
  diff_pool(bufS, 2000, 500, hA, 16, hB, adj0, 0, bufT, adjA, true);

  cheb_dense(hB, 500, 16, 16, adjA, 250000, enc_w[1], enc_b[1], hA);  relu(hA, (long)Bc * 500 * 16);
  cheb_dense(hA, 500, 16, 125, adjA, 250000, pool_w[1], pool_b[1], bufS);
  diff_pool(bufS, 500, 125, hA, 16, hB, adjA, 250000, bufT, adjB, true);

  cheb_dense(hB, 125, 16, 16, adjB, 15625, enc_w[2], enc_b[2], hA);   relu(hA, (long)Bc * 125 * 16);
  cheb_dense(hA, 125, 16, 32, adjB, 15625, pool_w[2], pool_b[2], bufS);
  diff_pool(bufS, 125, 32, hA, 16, hB, adjB, 15625, bufT, adjA, true);

  cheb_dense(hB, 32, 16, 32, adjA, 1024, enc_w[3], enc_b[3], hA);     relu(hA, (long)Bc * 32 * 32);
  cheb_dense(hA, 32, 32, 10, adjA, 1024, pool_w[3], pool_b[3], bufS);
  diff_pool(bufS, 32, 10, hA, 32, hB, adjA, 1024, bufT, adjB, true);  // hB=[B,10,32], adjB=[B,10,10]

  // ---------- latent ----------
  fillb(zb, enc_lin_b, Bc, 8);
  gemm(hB, enc_lin_w, zb, Bc, 8, 320, 320, 320, 8, 0, 0, 0, 1, 0, 1, 1.f, 1.f);
  relu(zb, Bc * 8);
  fillb(hA, dec_lin_b, Bc, 320);
  gemm(zb, dec_lin_w, hA, Bc, 320, 8, 8, 8, 320, 0, 0, 0, 1, 0, 1, 1.f, 1.f);
  relu(hA, (long)Bc * 320);                                  // hA = [B,10,32]

  // ---------- decoder ----------
  cheb_dense(hA, 10, 32, 32, adjB, 100, unp_w[0], unp_b[0], bufS);
  diff_pool(bufS, 10, 32, hA, 32, hB, adjB, 100, bufT, adjA, true);
  cheb_dense(hB, 32, 32, 16, adjA, 1024, dec_w[0], dec_b[0], hA);     relu(hA, (long)Bc * 32 * 16);

  cheb_dense(hA, 32, 16, 125, adjA, 1024, unp_w[1], unp_b[1], bufS);
  diff_pool(bufS, 32, 125, hA, 16, hB, adjA, 1024, bufT, adjB, true);
  cheb_dense(hB, 125, 16, 16, adjB, 15625, dec_w[1], dec_b[1], hA);   relu(hA, (long)Bc * 125 * 16);

  cheb_dense(hA, 125, 16, 500, adjB, 15625, unp_w[2], unp_b[2], bufS);
  diff_pool(bufS, 125, 500, hA, 16, hB, adjB, 15625, bufT, adjA, true);
  cheb_dense(hB, 500, 16, 16, adjA, 250000, dec_w[2], dec_b[2], hA);  relu(hA, (long)Bc * 500 * 16);

  cheb_dense(hA, 500, 16, 2000, adjA, 250000, unp_w[3], unp_b[3], bufS);
  diff_pool(bufS, 500, 2000, hA, 16, hB, adjA, 250000, bufT, (float*)0, false); // hB=[B,2000,16]; adj' dead

  float* dout = (float*)d_out;
  cheb_sparse(hB, 16, 3, dec_w[3], (const float*)0, dout);
  relu(dout, (long)Bc * N0 * 3);
  k_emit<<<1, 1, 0, stream>>>(dout, link, ent, (long)Bc * N0 * 3);
}